// SimpleLinearProbe_21706764714561
// MI455X (gfx1250) — compile-verified
//
#include <hip/hip_runtime.h>

typedef __attribute__((ext_vector_type(2))) float v2f;
typedef __attribute__((ext_vector_type(4))) float f4;
typedef __attribute__((ext_vector_type(8))) float v8f;

#define B_ 8
#define L_ 2048
#define H_ 768
#define R_ (B_ * L_)          // 16384 rows of E viewed as (R_, H_)

// ---------------------------------------------------------------------------
// Phase 1: [left | right] = E(16384 x 768) x W(768 x 2, padded to 16 cols)
// One wave32 per 16-row tile; V_WMMA_F32_16X16X4_F32 accumulating K=768/4.
//   A 16x4 f32 layout: lane m=lane&15 is row, VGPR0/1 hold K=(lane>>4)*2, +1.
//   B 4x16  f32 layout: lane n=lane&15 is column; only n==0 (W_left) and
//     n==1 (W_right) carry data -> built branch-free with v_cndmask.
//   C 16x16 f32 layout: VGPR r holds M=r (lanes 0-15) / M=r+8 (lanes 16-31).
// ---------------------------------------------------------------------------
__global__ __launch_bounds__(256) void probe_dot_wmma(
    const float* __restrict__ E,    // [R_, H_]
    const float* __restrict__ W,    // [2*H_]
    float* __restrict__ lr)         // [2*R_]  left then right
{
    const int lane  = threadIdx.x & 31;
    const int wave  = blockIdx.x * (blockDim.x >> 5) + (threadIdx.x >> 5);
    const int row0  = wave * 16;          // 1024 waves cover 16384 rows
    const int m     = lane & 15;          // A-row within tile (== B column n)
    const int khalf = (lane >> 4) * 2;    // K sub-offset: 0 or 2

    // Branch-free B operand setup (computed once):
    //   lanes n==1 read the right column; every other lane reads the left
    //   column (always a valid address) and is zeroed by a cndmask select.
    const bool  active = (m < 2);
    const float* ap = E + (size_t)(row0 + m) * H_ + khalf;   // A stream
    const float* wp = W + ((m == 1) ? H_ : 0) + khalf;        // B stream

    v8f c = {};
    #pragma unroll 4
    for (int k0 = 0; k0 < H_; k0 += 4) {
        v2f a;
        a.x = ap[k0];
        a.y = ap[k0 + 1];
        const float w0 = wp[k0];
        const float w1 = wp[k0 + 1];
        v2f b;
        b.x = active ? w0 : 0.0f;     // v_cndmask, mask hoisted out of loop
        b.y = active ? w1 : 0.0f;
        // D = A*B + C  (EXEC all-ones: no divergent control flow here)
        c = __builtin_amdgcn_wmma_f32_16x16x4_f32(
                /*neg_a=*/false, a, /*neg_b=*/false, b,
                /*c_mod=*/(short)0, c, /*reuse_a=*/false, /*reuse_b=*/false);
    }

    // Column 0 of C = left, column 1 = right.
    const int n     = lane & 15;
    const int mbase = (lane >> 4) * 8;
    if (n == 0) {
        #pragma unroll
        for (int r = 0; r < 8; ++r) lr[row0 + mbase + r] = c[r];
    } else if (n == 1) {
        #pragma unroll
        for (int r = 0; r < 8; ++r) lr[R_ + row0 + mbase + r] = c[r];
    }
}

// ---------------------------------------------------------------------------
// Phase 2: out[b,i,j] = left[b*L+i] + right[b*L+j] + bias.
// One float4 per thread -> global_store_b128; pure store-bandwidth bound
// (134 MB ~ 6 us at 23.3 TB/s). All index math is shifts/masks (L = 2^11).
// ---------------------------------------------------------------------------
__global__ __launch_bounds__(256) void probe_bcast_add(
    const float* __restrict__ lr,     // [2*R_]
    const float* __restrict__ bias,   // [1]
    float* __restrict__ out)          // [B_, L_, L_]
{
    const float bb = bias[0];
    const size_t t = (size_t)blockIdx.x * blockDim.x + threadIdx.x; // float4 id

    const int    j4   = (int)(t & (size_t)(L_ / 4 - 1));    // & 511
    const size_t rest = t >> 9;                              // / 512
    const int    i    = (int)(rest & (size_t)(L_ - 1));      // & 2047
    const int    b    = (int)(rest >> 11);                   // / 2048

    const float lv = lr[b * L_ + i] + bb;                    // uniform per wave
    const f4*   right4 = (const f4*)(lr + R_);
    const f4    r = right4[b * (L_ / 4) + j4];

    f4 o;
    o.x = lv + r.x;
    o.y = lv + r.y;
    o.z = lv + r.z;
    o.w = lv + r.w;
    ((f4*)out)[t] = o;
}

extern "C" void kernel_launch(void* const* d_in, const int* in_sizes, int n_in,
                              void* d_out, int out_size, void* d_ws, size_t ws_size,
                              hipStream_t stream) {
    const float* E    = (const float*)d_in[0];   // embeddings (8,2048,768) f32
    const float* W    = (const float*)d_in[1];   // (1536,) f32
    const float* bias = (const float*)d_in[2];   // (1,) f32
    float*       out  = (float*)d_out;           // (8,2048,2048) f32
    float*       lr   = (float*)d_ws;            // 2*16384 floats = 128 KB

    // Phase 1: 1024 waves = 128 blocks x 8 waves; each wave -> 16 rows.
    probe_dot_wmma<<<dim3(128), dim3(256), 0, stream>>>(E, W, lr);

    // Phase 2: 8*2048*2048/4 = 8,388,608 float4 stores.
    probe_bcast_add<<<dim3(32768), dim3(256), 0, stream>>>(lr, bias, out);
}